// EdgeModel_25666724561156
// MI455X (gfx1250) — compile-verified
//
#include <hip/hip_runtime.h>

// ---------------------------------------------------------------------------
// CDNA5 (gfx1250) fp32 WMMA implementation of the NNConv edge model.
// All matrix math uses V_WMMA_F32_16X16X4_F32 (exact fp32, matches reference).
// A-operands are sourced from bank-conflict-free padded LDS scratch so the
// lane-dependent K offset (lanes 16-31 take K+2) is pure address arithmetic.
// Full-tile fast paths keep stores/atomics branch-free (N, E multiples of 16).
// ---------------------------------------------------------------------------

typedef __attribute__((ext_vector_type(2))) float v2f;
typedef __attribute__((ext_vector_type(8))) float v8f;

#define RS  20  // scratch row stride (floats) for 16-wide rows: n*20 mod 64
                // covers all multiples of 4 -> wave ds_load_b64 hits all banks
#define RSP 36  // scratch row stride for 32-wide rows (same property)

__device__ __forceinline__ v8f wmma_k4(v2f a, v2f b, v8f c) {
  // D = A(16x4) * B(4x16) + C(16x16), fp32
  return __builtin_amdgcn_wmma_f32_16x16x4_f32(false, a, false, b, (short)0, c,
                                               false, false);
}

__device__ __forceinline__ v2f mkv2(float x, float y) {
  v2f v; v.x = x; v.y = y; return v;
}

// A-operand pair from LDS scratch: ds_load_b64 at lane-dependent address.
__device__ __forceinline__ v2f apair(const float* s) {
  float2 p = *(const float2*)s;
  return mkv2(p.x, p.y);
}

#define DS_FENCE() asm volatile("s_wait_dscnt 0" ::: "memory")

// ---------------------------------------------------------------------------
// Kernel 1/3: agg[m][:] = act(in[m][:]) @ R(16x16) + bias ; optionally also
// writes act(in) (relu) to hOut. One 16-row tile per wave, 4 WMMAs per tile.
// ---------------------------------------------------------------------------
__global__ __launch_bounds__(256) void node_transform_kernel(
    const float* __restrict__ xin, const float* __restrict__ R,
    const float* __restrict__ bias, float* __restrict__ aggOut,
    float* __restrict__ hOut, int N, int applyRelu) {
  __shared__ float Rl[256];
  __shared__ float bl[16];
  __shared__ float rS[8][16 * RS];

  for (int t = threadIdx.x; t < 256; t += blockDim.x) Rl[t] = R[t];
  if (threadIdx.x < 16) bl[threadIdx.x] = bias[threadIdx.x];
  __syncthreads();

  const int lane = threadIdx.x & 31;
  const int wave = threadIdx.x >> 5;
  const int hii = (lane >= 16) ? 1 : 0;
  const int n = lane & 15;
  float* rw = rS[wave];

  const int nTiles = (N + 15) >> 4;
  const int tile = blockIdx.x * 8 + wave;
  if (tile >= nTiles) return;  // wave-uniform: EXEC all-ones for WMMA
  const int base = tile * 16;
  const bool full = (base + 16 <= N);  // wave-uniform full-tile fast path
  const int m = base + n;
  const int mc = (m < N) ? m : (N - 1);

  if (lane < 16) {  // lanes 0-15 stage their row into LDS scratch
    const float4* p = (const float4*)(xin + (size_t)mc * 16);
#pragma unroll
    for (int j = 0; j < 4; ++j) {
      float4 v = p[j];
      if (applyRelu) {
        v.x = v.x > 0.f ? v.x : 0.f;
        v.y = v.y > 0.f ? v.y : 0.f;
        v.z = v.z > 0.f ? v.z : 0.f;
        v.w = v.w > 0.f ? v.w : 0.f;
      }
      *(float4*)&rw[n * RS + 4 * j] = v;
      if (applyRelu && hOut != nullptr && m < N)
        ((float4*)(hOut + (size_t)m * 16))[j] = v;
    }
  }
  DS_FENCE();

  v8f acc;
#pragma unroll
  for (int r = 0; r < 8; ++r) acc[r] = bl[n];

#pragma unroll
  for (int kk = 0; kk < 4; ++kk) {
    const int k0 = kk * 4;
    v2f a = apair(&rw[n * RS + k0 + 2 * hii]);
    v2f b = mkv2(Rl[(k0 + 2 * hii) * 16 + n], Rl[(k0 + 2 * hii + 1) * 16 + n]);
    acc = wmma_k4(a, b, acc);
  }

  if (full) {  // branch-free coalesced stores (common case)
#pragma unroll
    for (int r = 0; r < 8; ++r) {
      const int mo = base + r + 8 * hii;  // C-layout: vgpr r = row r (+8 hi)
      aggOut[(size_t)mo * 16 + n] = acc[r];
    }
  } else {
#pragma unroll
    for (int r = 0; r < 8; ++r) {
      const int mo = base + r + 8 * hii;
      if (mo < N) aggOut[(size_t)mo * 16 + n] = acc[r];
    }
  }
}

// ---------------------------------------------------------------------------
// Kernel 2/4: NNConv edge kernel.
//   t   = relu(edge_attr @ W1 + b1)                      (4 WMMAs / tile)
//   msg = [xs (x) t , xs] @ W2''(272x16)                 (68 WMMAs / tile)
//   atomicAdd msg into agg[dst]
// W2'' is W2 reordered ((i,k),o) <- (k, i*16+o) with b2 folded in as 16 extra
// K rows multiplied by xs. The outer product P[m][(i,k)] = xs[m][i]*t[m][k] is
// built per WMMA step: t-pair from LDS (ds_load_b64) scaled by register xs[i].
// ---------------------------------------------------------------------------
__global__ __launch_bounds__(256) void nnconv_edge_kernel(
    const float* __restrict__ xsrc, const float* __restrict__ eattr,
    const int* __restrict__ eidx, const float* __restrict__ W1,
    const float* __restrict__ b1, const float* __restrict__ W2,
    const float* __restrict__ b2, float* __restrict__ agg, int E, int N) {
  __shared__ float W1l[256];
  __shared__ float b1l[16];
  __shared__ float W2l[272 * 16];
  __shared__ float tS[8][16 * RS];  // per-wave: edge_attr rows, then t rows
  __shared__ float xS[8][16 * RS];  // per-wave: gathered x_src rows
  __shared__ int dS[8][16];         // per-wave: dst indices

  for (int t = threadIdx.x; t < 256; t += blockDim.x) W1l[t] = W1[t];
  if (threadIdx.x < 16) b1l[threadIdx.x] = b1[threadIdx.x];
  for (int t = threadIdx.x; t < 272 * 16; t += blockDim.x) {
    const int K = t >> 4, o = t & 15;
    if (K < 256) {
      const int i = K >> 4, k = K & 15;
      W2l[t] = W2[k * 256 + i * 16 + o];
    } else {
      W2l[t] = b2[(K - 256) * 16 + o];
    }
  }
  __syncthreads();

  const int lane = threadIdx.x & 31;
  const int wave = threadIdx.x >> 5;
  const int hii = (lane >= 16) ? 1 : 0;
  const int n = lane & 15;
  float* tw = tS[wave];
  float* xw = xS[wave];

  const int nTiles = (E + 15) >> 4;
  const int tile = blockIdx.x * 8 + wave;
  if (tile >= nTiles) return;
  const int base = tile * 16;
  const bool full = (base + 16 <= E);  // wave-uniform full-tile fast path
  const int e = base + n;
  const int ec = (e < E) ? e : (E - 1);

  const int src = eidx[ec];

  // xs row -> registers (outer-product multipliers; constant-indexed only)
  float xs[16];
  {
    const float4* p = (const float4*)(xsrc + (size_t)src * 16);
#pragma unroll
    for (int j = 0; j < 4; ++j) {
      float4 v = p[j];
      xs[4 * j + 0] = v.x; xs[4 * j + 1] = v.y;
      xs[4 * j + 2] = v.z; xs[4 * j + 3] = v.w;
    }
  }

  if (lane < 16) {  // stage xs rows, edge_attr rows, and dst into LDS
    dS[wave][n] = eidx[E + ec];
    const float4* pe = (const float4*)(eattr + (size_t)ec * 16);
#pragma unroll
    for (int j = 0; j < 4; ++j) {
      float4 v;
      v.x = xs[4 * j + 0]; v.y = xs[4 * j + 1];
      v.z = xs[4 * j + 2]; v.w = xs[4 * j + 3];
      *(float4*)&xw[n * RS + 4 * j] = v;
      *(float4*)&tw[n * RS + 4 * j] = pe[j];
    }
  }
  DS_FENCE();

  // ---- layer 1: t = relu(ea @ W1 + b1), result in C-layout ----
  v8f c1;
#pragma unroll
  for (int r = 0; r < 8; ++r) c1[r] = b1l[n];
#pragma unroll
  for (int kk = 0; kk < 4; ++kk) {
    const int k0 = kk * 4;
    v2f a = apair(&tw[n * RS + k0 + 2 * hii]);
    v2f b = mkv2(W1l[(k0 + 2 * hii) * 16 + n], W1l[(k0 + 2 * hii + 1) * 16 + n]);
    c1 = wmma_k4(a, b, c1);
  }
#pragma unroll
  for (int r = 0; r < 8; ++r) c1[r] = c1[r] > 0.f ? c1[r] : 0.f;

  // overwrite scratch with t rows (transpose via C-layout store)
#pragma unroll
  for (int r = 0; r < 8; ++r) tw[(r + 8 * hii) * RS + n] = c1[r];
  DS_FENCE();

  // ---- main GEMM: [16,272] @ [272,16] ----
  v8f acc;
#pragma unroll
  for (int r = 0; r < 8; ++r) acc[r] = 0.f;
#pragma unroll
  for (int kk = 0; kk < 68; ++kk) {
    const int k0 = kk * 4;
    v2f a;
    if (k0 < 256) {
      const int i = k0 >> 4, kb = k0 & 15;
      v2f tp = apair(&tw[n * RS + kb + 2 * hii]);
      const float xi = xs[i];
      a.x = xi * tp.x;
      a.y = xi * tp.y;
    } else {
      a = apair(&xw[n * RS + (k0 - 256) + 2 * hii]);
    }
    v2f b = mkv2(W2l[(k0 + 2 * hii) * 16 + n], W2l[(k0 + 2 * hii + 1) * 16 + n]);
    acc = wmma_k4(a, b, acc);
  }

  // ---- scatter-add messages to agg[dst] ----
  if (full) {  // branch-free atomics (common case)
#pragma unroll
    for (int r = 0; r < 8; ++r) {
      const int mr = r + 8 * hii;
      const int d = dS[wave][mr];  // broadcast ds_load
      atomicAdd(&agg[(size_t)d * 16 + n], acc[r]);
    }
  } else {
#pragma unroll
    for (int r = 0; r < 8; ++r) {
      const int mr = r + 8 * hii;
      const int d = dS[wave][mr];
      if (base + mr < E) atomicAdd(&agg[(size_t)d * 16 + n], acc[r]);
    }
  }
}

// ---------------------------------------------------------------------------
// Kernel 5: edge predictor MLP 32->32->16->16->1 over
// feat = [edge_attr, relu(agg2[src])]. 28 WMMAs / 16-edge tile; final 16->1
// as a per-lane row dot from LDS (no cross-lane shuffles).
// ---------------------------------------------------------------------------
__global__ __launch_bounds__(256) void predictor_kernel(
    const float* __restrict__ eattr, const int* __restrict__ eidx,
    const float* __restrict__ h2agg, const float* __restrict__ pW1,
    const float* __restrict__ pb1, const float* __restrict__ pW2,
    const float* __restrict__ pb2, const float* __restrict__ pW3,
    const float* __restrict__ pb3, const float* __restrict__ pW4,
    const float* __restrict__ pb4, float* __restrict__ out, int E, int N) {
  __shared__ float W1l[32 * 32];
  __shared__ float W2l[32 * 16];
  __shared__ float W3l[16 * 16];
  __shared__ float b1l[32], b2l[16], b3l[16];
  __shared__ float pS[8][16 * RSP];  // per-wave 32-wide row scratch

  for (int t = threadIdx.x; t < 1024; t += blockDim.x) W1l[t] = pW1[t];
  for (int t = threadIdx.x; t < 512; t += blockDim.x) W2l[t] = pW2[t];
  for (int t = threadIdx.x; t < 256; t += blockDim.x) W3l[t] = pW3[t];
  if (threadIdx.x < 32) b1l[threadIdx.x] = pb1[threadIdx.x];
  if (threadIdx.x < 16) {
    b2l[threadIdx.x] = pb2[threadIdx.x];
    b3l[threadIdx.x] = pb3[threadIdx.x];
  }
  __syncthreads();

  const int lane = threadIdx.x & 31;
  const int wave = threadIdx.x >> 5;
  const int hii = (lane >= 16) ? 1 : 0;
  const int n = lane & 15;
  float* pw = pS[wave];

  const int nTiles = (E + 15) >> 4;
  const int tile = blockIdx.x * 8 + wave;
  if (tile >= nTiles) return;
  const int base = tile * 16;
  const int e = base + n;
  const int ec = (e < E) ? e : (E - 1);

  if (lane < 16) {  // stage feat = [edge_attr, relu(agg2[src])] rows
    const int src = eidx[ec];
    const float4* pe = (const float4*)(eattr + (size_t)ec * 16);
    const float4* ph = (const float4*)(h2agg + (size_t)src * 16);
#pragma unroll
    for (int j = 0; j < 4; ++j) {
      *(float4*)&pw[n * RSP + 4 * j] = pe[j];
      float4 v = ph[j];
      v.x = v.x > 0.f ? v.x : 0.f;
      v.y = v.y > 0.f ? v.y : 0.f;
      v.z = v.z > 0.f ? v.z : 0.f;
      v.w = v.w > 0.f ? v.w : 0.f;
      *(float4*)&pw[n * RSP + 16 + 4 * j] = v;
    }
  }
  DS_FENCE();

  // ---- layer 1: [16,32] @ [32,32] (two N-chunks) ----
  v8f cA, cB;
#pragma unroll
  for (int r = 0; r < 8; ++r) { cA[r] = b1l[n]; cB[r] = b1l[16 + n]; }
#pragma unroll
  for (int kk = 0; kk < 8; ++kk) {
    const int k0 = kk * 4;
    v2f a = apair(&pw[n * RSP + k0 + 2 * hii]);
    v2f bA = mkv2(W1l[(k0 + 2 * hii) * 32 + n], W1l[(k0 + 2 * hii + 1) * 32 + n]);
    v2f bB = mkv2(W1l[(k0 + 2 * hii) * 32 + 16 + n],
                  W1l[(k0 + 2 * hii + 1) * 32 + 16 + n]);
    cA = wmma_k4(a, bA, cA);
    cB = wmma_k4(a, bB, cB);
  }
#pragma unroll
  for (int r = 0; r < 8; ++r) {
    cA[r] = cA[r] > 0.f ? cA[r] : 0.f;
    cB[r] = cB[r] > 0.f ? cB[r] : 0.f;
  }

  // z1 rows into scratch
#pragma unroll
  for (int r = 0; r < 8; ++r) {
    pw[(r + 8 * hii) * RSP + n] = cA[r];
    pw[(r + 8 * hii) * RSP + 16 + n] = cB[r];
  }
  DS_FENCE();

  // ---- layer 2: [16,32] @ [32,16] ----
  v8f c2;
#pragma unroll
  for (int r = 0; r < 8; ++r) c2[r] = b2l[n];
#pragma unroll
  for (int kk = 0; kk < 8; ++kk) {
    const int k0 = kk * 4;
    v2f a = apair(&pw[n * RSP + k0 + 2 * hii]);
    v2f b = mkv2(W2l[(k0 + 2 * hii) * 16 + n], W2l[(k0 + 2 * hii + 1) * 16 + n]);
    c2 = wmma_k4(a, b, c2);
  }
#pragma unroll
  for (int r = 0; r < 8; ++r) c2[r] = c2[r] > 0.f ? c2[r] : 0.f;

  // z2 rows into scratch (cols 0..15)
#pragma unroll
  for (int r = 0; r < 8; ++r) pw[(r + 8 * hii) * RSP + n] = c2[r];
  DS_FENCE();

  // ---- layer 3: [16,16] @ [16,16] ----
  v8f c3;
#pragma unroll
  for (int r = 0; r < 8; ++r) c3[r] = b3l[n];
#pragma unroll
  for (int kk = 0; kk < 4; ++kk) {
    const int k0 = kk * 4;
    v2f a = apair(&pw[n * RSP + k0 + 2 * hii]);
    v2f b = mkv2(W3l[(k0 + 2 * hii) * 16 + n], W3l[(k0 + 2 * hii + 1) * 16 + n]);
    c3 = wmma_k4(a, b, c3);
  }
#pragma unroll
  for (int r = 0; r < 8; ++r) c3[r] = c3[r] > 0.f ? c3[r] : 0.f;

  // z3 rows into scratch, then per-lane 16->1 dot (lanes 0-15)
#pragma unroll
  for (int r = 0; r < 8; ++r) pw[(r + 8 * hii) * RSP + n] = c3[r];
  DS_FENCE();

  if (lane < 16 && e < E) {
    float acc = pb4[0];
#pragma unroll
    for (int j = 0; j < 4; ++j) {
      float4 z = *(const float4*)&pw[n * RSP + 4 * j];
      float4 w = ((const float4*)pW4)[j];
      acc += z.x * w.x + z.y * w.y + z.z * w.z + z.w * w.w;
    }
    out[e] = acc;
  }
}

// ---------------------------------------------------------------------------
// Host orchestration: 5 launches, all on `stream`.
// ---------------------------------------------------------------------------
extern "C" void kernel_launch(void* const* d_in, const int* in_sizes, int n_in,
                              void* d_out, int out_size, void* d_ws,
                              size_t ws_size, hipStream_t stream) {
  const float* x       = (const float*)d_in[0];
  const float* eattr   = (const float*)d_in[1];
  const int*   eidx    = (const int*)d_in[2];
  const float* c1_W1   = (const float*)d_in[3];
  const float* c1_b1   = (const float*)d_in[4];
  const float* c1_W2   = (const float*)d_in[5];
  const float* c1_b2   = (const float*)d_in[6];
  const float* c1_root = (const float*)d_in[7];
  const float* c1_bias = (const float*)d_in[8];
  const float* c2_W1   = (const float*)d_in[9];
  const float* c2_b1   = (const float*)d_in[10];
  const float* c2_W2   = (const float*)d_in[11];
  const float* c2_b2   = (const float*)d_in[12];
  const float* c2_root = (const float*)d_in[13];
  const float* c2_bias = (const float*)d_in[14];
  const float* p_W1    = (const float*)d_in[15];
  const float* p_b1    = (const float*)d_in[16];
  const float* p_W2    = (const float*)d_in[17];
  const float* p_b2    = (const float*)d_in[18];
  const float* p_W3    = (const float*)d_in[19];
  const float* p_b3    = (const float*)d_in[20];
  const float* p_W4    = (const float*)d_in[21];
  const float* p_b4    = (const float*)d_in[22];

  const int N = in_sizes[0] / 16;
  const int E = in_sizes[2] / 2;

  float* agg1 = (float*)d_ws;           // [N,16]
  float* h1   = agg1 + (size_t)N * 16;  // [N,16]
  float* agg2 = h1 + (size_t)N * 16;    // [N,16]

  const int tilesN = (N + 15) / 16, blocksN = (tilesN + 7) / 8;
  const int tilesE = (E + 15) / 16, blocksE = (tilesE + 7) / 8;

  // 1) agg1 = x @ c1_root + c1_bias
  node_transform_kernel<<<blocksN, 256, 0, stream>>>(x, c1_root, c1_bias, agg1,
                                                     nullptr, N, 0);
  // 2) conv1 messages scatter-added into agg1
  nnconv_edge_kernel<<<blocksE, 256, 0, stream>>>(x, eattr, eidx, c1_W1, c1_b1,
                                                  c1_W2, c1_b2, agg1, E, N);
  // 3) h1 = relu(agg1); agg2 = h1 @ c2_root + c2_bias
  node_transform_kernel<<<blocksN, 256, 0, stream>>>(agg1, c2_root, c2_bias,
                                                     agg2, h1, N, 1);
  // 4) conv2 messages scatter-added into agg2
  nnconv_edge_kernel<<<blocksE, 256, 0, stream>>>(h1, eattr, eidx, c2_W1, c2_b1,
                                                  c2_W2, c2_b2, agg2, E, N);
  // 5) predictor over [edge_attr, relu(agg2[src])]
  predictor_kernel<<<blocksE, 256, 0, stream>>>(
      eattr, eidx, agg2, p_W1, p_b1, p_W2, p_b2, p_W3, p_b3, p_W4, p_b4,
      (float*)d_out, E, N);
}